// PolyConv_new_83657372991949
// MI455X (gfx1250) — compile-verified
//
#include <hip/hip_runtime.h>
#include <stdint.h>

constexpr int FEAT_DIM = 32;

// addrspacecast(LDS -> flat) keeps the workgroup-relative LDS byte offset in
// the low 32 bits (CDNA5 ISA 10.2 aperture mapping), which is exactly what the
// async-to-LDS VDST operand wants (a byte offset added to the wave's LDS base).
__device__ __forceinline__ uint32_t lds_offset_of(const void* p) {
  return (uint32_t)(uintptr_t)p;
}

// CDNA5 async copy global -> LDS, tracked by ASYNCcnt.
__device__ __forceinline__ void async_load_b32_to_lds(uint32_t lds_byte_off,
                                                      const void* gaddr) {
  asm volatile("global_load_async_to_lds_b32 %0, %1, off"
               :
               : "v"(lds_byte_off), "v"(gaddr)
               : "memory");
}

__device__ __forceinline__ void wait_asynccnt0() {
  asm volatile("s_wait_asynccnt 0" ::: "memory");
}

__global__ void zero_kernel(float4* __restrict__ p, int n4) {
  int i = blockIdx.x * blockDim.x + threadIdx.x;
  if (i < n4) p[i] = make_float4(0.f, 0.f, 0.f, 0.f);
}

// deg[dst] += mask  (edge data staged through LDS with async loads)
__global__ void degree_kernel(const int* __restrict__ dst,
                              const float* __restrict__ mask,
                              float* __restrict__ deg, int n_edges) {
  __shared__ int   s_dst[256];
  __shared__ float s_mask[256];
  const int t = threadIdx.x;
  const int e = blockIdx.x * 256 + t;
  if (e < n_edges) {
    async_load_b32_to_lds(lds_offset_of(&s_dst[t]),  dst  + e);
    async_load_b32_to_lds(lds_offset_of(&s_mask[t]), mask + e);
  }
  wait_asynccnt0();
  __syncthreads();
  if (e < n_edges) {
    unsafeAtomicAdd(&deg[s_dst[t]], s_mask[t]);
  }
}

__global__ void dinv_kernel(const float* __restrict__ deg,
                            float* __restrict__ dinv, int n) {
  int i = blockIdx.x * blockDim.x + threadIdx.x;
  if (i < n) dinv[i] = rsqrtf(fmaxf(deg[i], 1.0f));
}

// h = theta0 * feat ; feat_cur = feat   (float4 vectorized)
__global__ void init_kernel(const float4* __restrict__ feat,
                            float4* __restrict__ feat_cur,
                            float4* __restrict__ h, float theta0, int n4) {
  int i = blockIdx.x * blockDim.x + threadIdx.x;
  if (i < n4) {
    float4 f = feat[i];
    feat_cur[i] = f;
    h[i] = make_float4(theta0 * f.x, theta0 * f.y, theta0 * f.z, theta0 * f.w);
  }
}

// agg[dst] += feat_cur[src] * (dinv[src] * mask)
// 32 edges per 256-thread block (8 waves). Wave 0 stages src/dst/mask via
// async LDS loads and folds dinv[src]*mask into a per-edge weight; then each
// edge is processed by 8 lanes, one float4 (16B) of the 128B feature row each,
// so a wave's 32 lanes issue fully-coalesced row gathers and 32 independent
// f32 atomics (L2-resident agg).
__global__ void edge_kernel(const int* __restrict__ src,
                            const int* __restrict__ dst,
                            const float* __restrict__ mask,
                            const float* __restrict__ dinv,
                            const float* __restrict__ feat_cur,
                            float* __restrict__ agg, int n_edges) {
  __shared__ int   s_src[32];
  __shared__ int   s_dst[32];
  __shared__ float s_w[32];

  const int t  = threadIdx.x;        // 0..255
  const int eb = blockIdx.x * 32;    // first edge of this block

  if (t < 32) {                      // exactly wave 0 (wave32)
    const int e = eb + t;
    if (e < n_edges) {
      async_load_b32_to_lds(lds_offset_of(&s_src[t]), src  + e);
      async_load_b32_to_lds(lds_offset_of(&s_dst[t]), dst  + e);
      async_load_b32_to_lds(lds_offset_of(&s_w[t]),   mask + e);
    }
  }
  wait_asynccnt0();
  __syncthreads();
  if (t < 32) {
    const int e = eb + t;
    if (e < n_edges) s_w[t] *= dinv[s_src[t]];   // w = mask * d^{-1/2}[src]
  }
  __syncthreads();

  const int le = t >> 3;             // local edge 0..31
  const int e  = eb + le;
  if (e >= n_edges) return;
  const int c  = (t & 7) << 2;       // feature base: 0,4,...,28

  const float  w = s_w[le];
  const float4 f =
      *reinterpret_cast<const float4*>(feat_cur + (size_t)s_src[le] * FEAT_DIM + c);
  float* a = agg + (size_t)s_dst[le] * FEAT_DIM + c;
  unsafeAtomicAdd(a + 0, f.x * w);
  unsafeAtomicAdd(a + 1, f.y * w);
  unsafeAtomicAdd(a + 2, f.z * w);
  unsafeAtomicAdd(a + 3, f.w * w);
}

// feat_cur -= agg * dinv ; h += theta * feat_cur ; agg = 0 (for next iter).
// On the last polynomial term, feat_cur/agg are dead -> skip those stores.
template <bool LAST>
__global__ void update_kernel(float4* __restrict__ feat_cur,
                              float4* __restrict__ agg,
                              float4* __restrict__ h,
                              const float* __restrict__ dinv,
                              float theta, int n4) {
  int i = blockIdx.x * blockDim.x + threadIdx.x;
  if (i >= n4) return;
  const float di = dinv[i >> 3];     // 8 float4 per node (32 feats)
  float4 f = feat_cur[i];
  float4 a = agg[i];
  f.x = fmaf(-a.x, di, f.x);
  f.y = fmaf(-a.y, di, f.y);
  f.z = fmaf(-a.z, di, f.z);
  f.w = fmaf(-a.w, di, f.w);
  if (!LAST) {
    feat_cur[i] = f;
    agg[i] = make_float4(0.f, 0.f, 0.f, 0.f);
  }
  float4 hh = h[i];
  hh.x = fmaf(theta, f.x, hh.x);
  hh.y = fmaf(theta, f.y, hh.y);
  hh.z = fmaf(theta, f.z, hh.z);
  hh.w = fmaf(theta, f.w, hh.w);
  h[i] = hh;
}

extern "C" void kernel_launch(void* const* d_in, const int* in_sizes, int n_in,
                              void* d_out, int out_size, void* d_ws, size_t ws_size,
                              hipStream_t stream) {
  const float* feat = (const float*)d_in[0];
  const int*   src  = (const int*)d_in[1];
  const int*   dst  = (const int*)d_in[2];
  const float* mask = (const float*)d_in[3];

  const int n_nodes = in_sizes[0] / FEAT_DIM;   // 100000
  const int n_edges = in_sizes[1];              // 1600000

  // Workspace layout (all 16B aligned for n_nodes=100000):
  // [deg: N][dinv: N][feat_cur: N*32][agg: N*32]  -> ~26.4 MB, L2-resident.
  float* ws       = (float*)d_ws;
  float* deg      = ws;
  float* dinv     = ws + n_nodes;
  float* feat_cur = ws + 2 * (size_t)n_nodes;
  float* agg      = feat_cur + (size_t)n_nodes * FEAT_DIM;
  float* h        = (float*)d_out;

  const float theta[5] = {0.2f, -0.4f, 0.3f, -0.15f, 0.05f};

  const int n4      = (n_nodes * FEAT_DIM) / 4;   // 800000 float4
  const int n4_deg  = (n_nodes + 3) / 4;
  const int B       = 256;

  // Fresh per-call state (ws is poisoned by harness; re-init every call).
  zero_kernel<<<(n4_deg + B - 1) / B, B, 0, stream>>>((float4*)deg, n4_deg);
  zero_kernel<<<(n4 + B - 1) / B, B, 0, stream>>>((float4*)agg, n4);

  degree_kernel<<<(n_edges + B - 1) / B, B, 0, stream>>>(dst, mask, deg, n_edges);
  dinv_kernel<<<(n_nodes + B - 1) / B, B, 0, stream>>>(deg, dinv, n_nodes);
  init_kernel<<<(n4 + B - 1) / B, B, 0, stream>>>((const float4*)feat,
                                                  (float4*)feat_cur, (float4*)h,
                                                  theta[0], n4);

  const int eblocks = (n_edges + 31) / 32;   // 32 edges per block
  const int ublocks = (n4 + B - 1) / B;
  for (int k = 1; k < 5; ++k) {
    edge_kernel<<<eblocks, B, 0, stream>>>(src, dst, mask, dinv, feat_cur, agg,
                                           n_edges);
    if (k < 4) {
      update_kernel<false><<<ublocks, B, 0, stream>>>(
          (float4*)feat_cur, (float4*)agg, (float4*)h, dinv, theta[k], n4);
    } else {
      update_kernel<true><<<ublocks, B, 0, stream>>>(
          (float4*)feat_cur, (float4*)agg, (float4*)h, dinv, theta[k], n4);
    }
  }
}